// HeteroResGatedGraphConvLayer_82308753260704
// MI455X (gfx1250) — compile-verified
//
#include <hip/hip_runtime.h>

// HeteroResGatedGraphConv for MI455X (gfx1250, wave32, WMMA).
//
// out_i = (1/T) * sum_{e: dst=i} sigmoid(K[t_e][i] + Q[t_e][src_e]) * V[t_e][src_e]
//       + x_i @ mean_t(Ws[t]) + mean_t(b[t])
//
// Phase 1: 12 fp32 GEMMs (K/Q/V per type) via V_WMMA_F32_16X16X4_F32 -> d_ws
// Phase 2: averaged-Ws GEMM -> d_out (dense/residual part, overwrites poison)
// Phase 3: edge scatter with global f32 atomics into d_out

typedef __attribute__((ext_vector_type(2))) float v2f;
typedef __attribute__((ext_vector_type(8))) float v8f;

#define NN 50000   // nodes  (divisible by 16 -> all WMMA row-tiles full)
#define NE 800000  // edges
#define DF 128     // feature dim (in == out)
#define NT 4       // edge types

// ---------------------------------------------------------------------------
// GEMM: Y[row][col] = sum_k X[row][k] * W[k][col] + bias[col]
// Block: 256 threads = 8 waves; each wave -> one 16x128 output tile.
// W^T staged in LDS in two K-halves (128 x 68 floats, padded stride).
// ---------------------------------------------------------------------------
__global__ __launch_bounds__(256) void gemm_x_w_kernel(
    const float* __restrict__ X, const float* __restrict__ W,
    const float* __restrict__ bias, float* __restrict__ Y) {
  __shared__ float ldsW[128 * 68];  // ldsW[n*68 + kk] = W[h*64+kk][n], 34.8 KB

  const int tid  = threadIdx.x;
  const int wave = tid >> 5;
  const int lane = tid & 31;
  const int lrow = lane & 15;   // row (A) / col (B) index within 16
  const int ksel = lane >> 4;   // 0 -> K{0,1}, 1 -> K{2,3} within a 4-step
  const long row0 = (long)blockIdx.x * 128 + (long)wave * 16;
  const bool active = (row0 < NN);  // wave-uniform: EXEC all-ones under it

  v8f acc[8] = {};  // 8 x (16x16 f32 C/D tiles) = columns 0..127

  for (int h = 0; h < 2; ++h) {
    // Stage transposed half of W: kk in [0,64)
    for (int i = tid; i < 128 * 64; i += 256) {
      const int kk = i >> 7;    // 0..63
      const int n  = i & 127;   // 0..127
      ldsW[n * 68 + kk] = W[(h * 64 + kk) * DF + n];
    }
    __syncthreads();

    if (active) {
      // A fragment source: lane holds X[row0+lrow][k0 + ksel*2 .. +1]
      const float* xrow = X + (row0 + lrow) * DF + h * 64 + ksel * 2;
#pragma unroll
      for (int kb = 0; kb < 16; ++kb) {      // k0 = h*64 + kb*4
        const v2f a = *(const v2f*)(xrow + kb * 4);
#pragma unroll
        for (int nt = 0; nt < 8; ++nt) {     // column tile
          // B fragment: lane holds W[k0 + ksel*2 .. +1][nt*16 + lrow]
          const v2f b =
              *(const v2f*)(&ldsW[(nt * 16 + lrow) * 68 + kb * 4 + ksel * 2]);
          acc[nt] = __builtin_amdgcn_wmma_f32_16x16x4_f32(
              false, a, false, b, (short)0, acc[nt], false, false);
        }
      }
    }
    __syncthreads();
  }

  if (active) {
    // C/D layout: vgpr r, lanes 0-15 -> (M=r, N=lane); lanes 16-31 -> (M=8+r)
#pragma unroll
    for (int nt = 0; nt < 8; ++nt) {
      const int col = nt * 16 + lrow;
      const float bv = bias[col];
#pragma unroll
      for (int r = 0; r < 8; ++r) {
        Y[(row0 + ksel * 8 + r) * DF + col] = acc[nt][r] + bv;
      }
    }
  }
}

// ---------------------------------------------------------------------------
// Fold the residual projections: Wavg = mean_t Ws[t], bavg = mean_t b[t]
// ---------------------------------------------------------------------------
__global__ void avg_ws_kernel(const float* __restrict__ Ws,
                              const float* __restrict__ b,
                              float* __restrict__ Wavg,
                              float* __restrict__ bavg) {
  const int i = blockIdx.x * blockDim.x + threadIdx.x;
  if (i < DF * DF) {
    Wavg[i] = 0.25f * (Ws[i] + Ws[i + DF * DF] + Ws[i + 2 * DF * DF] +
                       Ws[i + 3 * DF * DF]);
  }
  if (i < DF) {
    bavg[i] = 0.25f * (b[i] + b[i + DF] + b[i + 2 * DF] + b[i + 3 * DF]);
  }
}

// ---------------------------------------------------------------------------
// Edge scatter: 128 threads per edge (one per channel), 2 edges in flight
// per 256-thread block, EPB edges per block total.
// ---------------------------------------------------------------------------
#define EPB 8

__global__ __launch_bounds__(256) void edge_scatter_kernel(
    const float* __restrict__ Kb, const float* __restrict__ Qb,
    const float* __restrict__ Vb, const int* __restrict__ src,
    const int* __restrict__ dst, const int* __restrict__ etype,
    float* __restrict__ out) {
  const int c   = threadIdx.x & 127;  // channel
  const int sub = threadIdx.x >> 7;   // 0..1
  const long e0 = (long)blockIdx.x * EPB;
#pragma unroll
  for (int it = 0; it < EPB / 2; ++it) {
    const long e = e0 + (long)it * 2 + sub;
    if (e < NE) {
      const int t = etype[e];
      const int s = src[e];
      const int d = dst[e];
      const long ki = ((long)t * NN + d) * DF + c;
      const long qi = ((long)t * NN + s) * DF + c;
      const float kv = Kb[ki];
      const float qv = Qb[qi];
      const float vv = Vb[qi];
      const float eta = 1.0f / (1.0f + __expf(-(kv + qv)));
      atomicAdd(&out[(long)d * DF + c], 0.25f * eta * vv);
    }
  }
}

// ---------------------------------------------------------------------------
extern "C" void kernel_launch(void* const* d_in, const int* in_sizes, int n_in,
                              void* d_out, int out_size, void* d_ws,
                              size_t ws_size, hipStream_t stream) {
  const float* x    = (const float*)d_in[0];
  const int* eidx   = (const int*)d_in[1];   // [2][E]: row0=src, row1=dst
  const int* etype  = (const int*)d_in[2];
  const float* Wk   = (const float*)d_in[3];
  const float* bk   = (const float*)d_in[4];
  const float* Wq   = (const float*)d_in[5];
  const float* bq   = (const float*)d_in[6];
  const float* Wv   = (const float*)d_in[7];
  const float* bv   = (const float*)d_in[8];
  const float* Ws   = (const float*)d_in[9];
  const float* bb   = (const float*)d_in[10];
  float* out = (float*)d_out;
  float* ws  = (float*)d_ws;

  const size_t PER = (size_t)NT * NN * DF;  // 25.6M floats per K/Q/V bank
  float* Kb   = ws;
  float* Qb   = ws + PER;
  float* Vb   = ws + 2 * PER;
  float* Wavg = ws + 3 * PER;
  float* bavg = Wavg + DF * DF;

  const dim3 gblk((NN + 127) / 128);  // 391 row-blocks
  const dim3 tblk(256);

  // Phase 1: K/Q/V projections per edge type (fp32 WMMA GEMMs)
  for (int t = 0; t < NT; ++t) {
    gemm_x_w_kernel<<<gblk, tblk, 0, stream>>>(
        x, Wk + (size_t)t * DF * DF, bk + (size_t)t * DF,
        Kb + (size_t)t * NN * DF);
    gemm_x_w_kernel<<<gblk, tblk, 0, stream>>>(
        x, Wq + (size_t)t * DF * DF, bq + (size_t)t * DF,
        Qb + (size_t)t * NN * DF);
    gemm_x_w_kernel<<<gblk, tblk, 0, stream>>>(
        x, Wv + (size_t)t * DF * DF, bv + (size_t)t * DF,
        Vb + (size_t)t * NN * DF);
  }

  // Phase 2: averaged residual projection straight into d_out
  avg_ws_kernel<<<dim3(64), tblk, 0, stream>>>(Ws, bb, Wavg, bavg);
  gemm_x_w_kernel<<<gblk, tblk, 0, stream>>>(x, Wavg, bavg, out);

  // Phase 3: gated message scatter (atomic f32 add into out)
  const int nblk = (NE + EPB - 1) / EPB;
  edge_scatter_kernel<<<dim3(nblk), tblk, 0, stream>>>(
      Kb, Qb, Vb, eidx, eidx + NE, etype, out);
}